// RNN_20968030339588
// MI455X (gfx1250) — compile-verified
//
#include <hip/hip_runtime.h>
#include <hip/hip_bf16.h>

typedef float v2f __attribute__((ext_vector_type(2)));
typedef float v8f __attribute__((ext_vector_type(8)));

#define T_STEPS 128
#define BATCH   64
#define DIM     1024
#define HID     1024
#define G4      4096   // 4*HID

// --------------------------------------------------------------------------
// CDNA5 async global->LDS copy (ASYNCcnt path), with guaranteed-compile
// fallback to synchronous float4 staging if the builtins are absent.
// Builtin signature (confirmed by round-2 diagnostic):
//   void __builtin_amdgcn_global_load_async_to_lds_b128(
//       v4i __device__(AS1)* gsrc, v4i __shared__(AS3)* ldst,
//       imm int offset, imm int cpol)
// --------------------------------------------------------------------------
#if __has_builtin(__builtin_amdgcn_global_load_async_to_lds_b128) && \
    __has_builtin(__builtin_amdgcn_s_wait_asynccnt)
#define HAVE_ASYNC_LDS 1
#else
#define HAVE_ASYNC_LDS 0
#endif

typedef int v4i __attribute__((__vector_size__(16)));
typedef __attribute__((address_space(1))) v4i as1_v4i;
typedef __attribute__((address_space(3))) v4i as3_v4i;

__device__ __forceinline__ void lds_copy16(float* lds_dst, const float* gsrc) {
#if HAVE_ASYNC_LDS
  __builtin_amdgcn_global_load_async_to_lds_b128(
      (as1_v4i*)const_cast<float*>(gsrc), (as3_v4i*)lds_dst, 0, 0);
#else
  *(float4*)lds_dst = *(const float4*)gsrc;
#endif
}

// wait until at most N async copies of this wave remain in flight
// (async loads complete in order, so <=N pending => older copies landed)
#if HAVE_ASYNC_LDS
#define ASYNC_WAIT(n) __builtin_amdgcn_s_wait_asynccnt(n)
#else
#define ASYNC_WAIT(n) do {} while (0)
#endif

// ---------------------------------------------------------------------------
// zero-init helper (h0 / h1 / c state)
// ---------------------------------------------------------------------------
__global__ void hlstm_zero(float* __restrict__ p, int n) {
  int i = blockIdx.x * blockDim.x + threadIdx.x;
  if (i < n) p[i] = 0.0f;
}

// ---------------------------------------------------------------------------
// Kernel 1: pre-gates  G[m][n] = X[m][:] @ Wx[:][n] + bias[n]
//   M = T*B = 8192, N = 4H = 4096, K = D = 1024
//   block = 256 threads (8 waves), block tile 32(M) x 64(N), KT = 32
//   double-buffered LDS: async copies of tile k+1 overlap WMMA on tile k
// ---------------------------------------------------------------------------
__global__ __launch_bounds__(256) void hlstm_pregemm(
    const float* __restrict__ X, const float* __restrict__ Wx,
    const float* __restrict__ bias, float* __restrict__ G)
{
  __shared__ float As[2 * 32 * 36];   // 2 x (32x32 tile, padded stride 36)
  __shared__ float Bs[2 * 32 * 64];   // 2 x (32x64 tile)

  const int tid  = threadIdx.x;
  const int wave = tid >> 5;
  const int lane = tid & 31;
  const int half = lane >> 4;     // 0: K0/K1 half, 1: K2/K3 half
  const int l16  = lane & 15;

  const int mTile = wave >> 2;
  const int nTile = wave & 3;
  const int gm = blockIdx.y * 32;
  const int gn = blockIdx.x * 64;

  v8f acc;
  const float bval = bias[gn + nTile * 16 + l16];
  #pragma unroll
  for (int r = 0; r < 8; ++r) acc[r] = bval;

  const int arow = tid >> 3, acg = (tid & 7) << 2;   // A: 32 rows x 8 float4
  const int brow = tid >> 4, bcg = (tid & 15) << 2;  // B: 16 rows x 16 float4 (x2)

  const float* gA  = &X [(size_t)(gm + arow) * DIM + acg];        // + kk
  const float* gB0 = &Wx[(size_t)brow        * G4  + gn + bcg];   // + kk*G4
  const float* gB1 = &Wx[(size_t)(brow + 16) * G4  + gn + bcg];   // + kk*G4
  const int aoff  = arow * 36 + acg;
  const int b0off = brow * 64 + bcg;
  const int b1off = (brow + 16) * 64 + bcg;

  // prologue: stage first K tile into buffer 0
  lds_copy16(&As[aoff],  gA);
  lds_copy16(&Bs[b0off], gB0);
  lds_copy16(&Bs[b1off], gB1);

  int buf = 0;
  for (int kk = 0; kk < DIM; kk += 32, buf ^= 1) {
    const int nb = buf ^ 1;
    if (kk + 32 < DIM) {   // issue next tile, then wait for current (3 pending)
      lds_copy16(&As[nb * 1152 + aoff],  gA  + (kk + 32));
      lds_copy16(&Bs[nb * 2048 + b0off], gB0 + (size_t)(kk + 32) * G4);
      lds_copy16(&Bs[nb * 2048 + b1off], gB1 + (size_t)(kk + 32) * G4);
      ASYNC_WAIT(3);
    } else {
      ASYNC_WAIT(0);
    }
    __syncthreads();   // all waves' current-buffer copies visible

    const float* A = &As[buf * 1152];
    const float* B = &Bs[buf * 2048];
    #pragma unroll
    for (int k4 = 0; k4 < 32; k4 += 4) {
      v2f a, b;
      const int ka = k4 + 2 * half;
      a.x = A[(mTile * 16 + l16) * 36 + ka];
      a.y = A[(mTile * 16 + l16) * 36 + ka + 1];
      b.x = B[ka * 64 + nTile * 16 + l16];
      b.y = B[(ka + 1) * 64 + nTile * 16 + l16];
      acc = __builtin_amdgcn_wmma_f32_16x16x4_f32(
          false, a, false, b, (short)0, acc, false, false);
    }
    __syncthreads();   // current buffer reusable for iteration kk+64
  }

  #pragma unroll
  for (int r = 0; r < 8; ++r)
    G[(size_t)(gm + mTile * 16 + r + 8 * half) * G4 + gn + nTile * 16 + l16] = acc[r];
}

// ---------------------------------------------------------------------------
// Kernel 2: one recurrence step (launched T times, stream-ordered)
//   gates = Gt + Hprev @ Wh ; hard activations ; c,h update ; emit h
//   grid.x = HID/16 = 64 blocks, block = 512 threads (16 waves)
//   wave w -> (gate = w/4, mTile = w%4); block owns h-cols [n0, n0+16)
//   for ALL four gates so the activation fuses locally. Double-buffered LDS.
// ---------------------------------------------------------------------------
__global__ __launch_bounds__(512) void hlstm_step(
    const float* __restrict__ Gt,    // [BATCH, 4H] pre-gates for this step
    const float* __restrict__ Wh,    // [HID, 4H]
    const float* __restrict__ Hprev, // [BATCH, HID]
    float* __restrict__ Hnew,        // [BATCH, HID]
    float* __restrict__ C,           // [BATCH, HID], updated in place
    float* __restrict__ Out)         // d_out + t*B*H
{
  __shared__ float As[2 * 64 * 36];  // 2 x (Hprev tile 64x32, pad 36)
  __shared__ float Bs[2 * 32 * 64];  // 2 x (Wh tile 32 x (4 gates x 16))
  __shared__ float Gs[64 * 68];      // raw gates, stride 68 = conflict-free

  const int tid  = threadIdx.x;
  const int wave = tid >> 5;       // 0..15
  const int lane = tid & 31;
  const int half = lane >> 4;
  const int l16  = lane & 15;
  const int gate  = wave >> 2;     // 0..3 : i,f,g,o
  const int mTile = wave & 3;      // 0..3 : batch 16-row tile
  const int n0 = blockIdx.x * 16;  // h column base

  // accumulator initialized from precomputed x@Wx + b pre-gates
  v8f acc;
  #pragma unroll
  for (int r = 0; r < 8; ++r)
    acc[r] = Gt[(size_t)(mTile * 16 + r + 8 * half) * G4 + gate * HID + n0 + l16];

  const int arow = tid >> 3, acg = (tid & 7) << 2;   // A: 64 rows x 8 float4
  const int brow = tid >> 4;                          // B: 32 rows x 16 float4
  const int bq = tid & 15;
  const int bg = bq >> 2, bj = (bq & 3) << 2;

  const float* gA = &Hprev[(size_t)arow * HID + acg];                 // + kk
  const float* gB = &Wh[(size_t)brow * G4 + bg * HID + n0 + bj];      // + kk*G4
  const int aoff = arow * 36 + acg;
  const int boff = brow * 64 + bg * 16 + bj;

  // prologue: stage first K tile into buffer 0
  lds_copy16(&As[aoff], gA);
  lds_copy16(&Bs[boff], gB);

  int buf = 0;
  for (int kk = 0; kk < HID; kk += 32, buf ^= 1) {
    const int nb = buf ^ 1;
    if (kk + 32 < HID) {   // issue next tile, wait for current (2 pending)
      lds_copy16(&As[nb * 2304 + aoff], gA + (kk + 32));
      lds_copy16(&Bs[nb * 2048 + boff], gB + (size_t)(kk + 32) * G4);
      ASYNC_WAIT(2);
    } else {
      ASYNC_WAIT(0);
    }
    __syncthreads();

    const float* A = &As[buf * 2304];
    const float* B = &Bs[buf * 2048];
    #pragma unroll
    for (int k4 = 0; k4 < 32; k4 += 4) {
      v2f a, b;
      const int ka = k4 + 2 * half;
      a.x = A[(mTile * 16 + l16) * 36 + ka];
      a.y = A[(mTile * 16 + l16) * 36 + ka + 1];
      b.x = B[ka * 64 + gate * 16 + l16];
      b.y = B[(ka + 1) * 64 + gate * 16 + l16];
      acc = __builtin_amdgcn_wmma_f32_16x16x4_f32(
          false, a, false, b, (short)0, acc, false, false);
    }
    __syncthreads();
  }

  // stage raw gate tiles for the fused elementwise phase (stride 68: the
  // lanes>=16 half (M+8) lands on banks shifted by 32 -> conflict-free)
  #pragma unroll
  for (int r = 0; r < 8; ++r)
    Gs[(mTile * 16 + r + 8 * half) * 68 + gate * 16 + l16] = acc[r];
  __syncthreads();

  // elementwise: 64 batch x 16 h-cols = 1024 outputs, 2 per thread
  #pragma unroll
  for (int e = 0; e < 2; ++e) {
    const int idx = tid + e * 512;
    const int row = idx >> 4, j = idx & 15;
    float iv = Gs[row * 68 + j];
    float fv = Gs[row * 68 + 16 + j];
    float gv = Gs[row * 68 + 32 + j];
    float ov = Gs[row * 68 + 48 + j];
    iv = fminf(fmaxf(iv + 0.5f, 0.0f), 1.0f);   // hard_sigmoid
    fv = fminf(fmaxf(fv + 0.5f, 0.0f), 1.0f);
    gv = fminf(fmaxf(gv, -1.0f), 1.0f);         // hard_tanh
    ov = fminf(fmaxf(ov + 0.5f, 0.0f), 1.0f);
    const size_t cidx = (size_t)row * HID + n0 + j;
    const float cn = fv * C[cidx] + iv * gv;
    C[cidx] = cn;
    const float hv = ov * fminf(fmaxf(cn, -1.0f), 1.0f);
    Hnew[cidx] = hv;
    Out[cidx]  = hv;
  }
}

// ---------------------------------------------------------------------------
// launch
// ---------------------------------------------------------------------------
extern "C" void kernel_launch(void* const* d_in, const int* in_sizes, int n_in,
                              void* d_out, int out_size, void* d_ws, size_t ws_size,
                              hipStream_t stream) {
  (void)in_sizes; (void)n_in; (void)out_size; (void)ws_size;

  const float* x    = (const float*)d_in[0];  // [T, B, D]
  const float* Wx   = (const float*)d_in[1];  // [D, 4H]
  const float* Wh   = (const float*)d_in[2];  // [H, 4H]
  const float* bias = (const float*)d_in[3];  // [4H]
  float* out = (float*)d_out;                 // [T, B, H]

  // workspace layout: G | h0 | h1 | c
  float* ws = (float*)d_ws;
  float* G  = ws;                                     // T*B*4H = 128 MB
  const size_t Gn = (size_t)T_STEPS * BATCH * G4;
  float* h0 = G + Gn;
  float* h1 = h0 + (size_t)BATCH * HID;
  float* c  = h1 + (size_t)BATCH * HID;

  // zero h0, h1, c (contiguous 3*B*H floats)
  {
    const int zn = 3 * BATCH * HID;
    hlstm_zero<<<(zn + 255) / 256, 256, 0, stream>>>(h0, zn);
  }

  // big parallel GEMM: all input projections at once
  {
    dim3 grid(G4 / 64, (T_STEPS * BATCH) / 32);  // (64, 256)
    hlstm_pregemm<<<grid, 256, 0, stream>>>(x, Wx, bias, G);
  }

  // sequential recurrence, stream-ordered; h ping-pongs between h0/h1
  for (int t = 0; t < T_STEPS; ++t) {
    const float* hp = (t & 1) ? h1 : h0;
    float*       hn = (t & 1) ? h0 : h1;
    hlstm_step<<<dim3(HID / 16), 512, 0, stream>>>(
        G + (size_t)t * BATCH * G4, Wh, hp, hn, c,
        out + (size_t)t * BATCH * HID);
  }
}